// Model_5454608466250
// MI455X (gfx1250) — compile-verified
//
#include <hip/hip_runtime.h>
#include <math.h>

// ---------------- problem constants ----------------
#define BN        8          // B*E events
#define BATCH     4
#define MODEL_DIM 16
#define NFRAMES   128
#define NTAB      4
#define TSIZE     512
#define S_TOT     32768
#define HOP       256
#define INV_GAUSS 39.89422804014327f   // 1/(0.01*sqrt(2*pi))

// d_out layout: x (4*1*32768) then sk (8*32768*512)
#define X_OFF   0
#define SK_OFF  131072

// workspace layout (float offsets)
#define C16_OFF   0          // (8,4,16) softmaxed choice anchors
#define ENVF_OFF  512        // (8,128)  |env| frames
#define FREQF_OFF 1536       // (8,128)  freq frames (post-sigmoid)
#define WTMAX_OFF 2560       // (4)      per-table max
#define PREF_OFF  2564       // (8,128)  phase block prefixes
#define MIX_OFF   4096       // (8,16,512) mixed table anchors, j-major
#define EV_OFF    69632      // (8,32768) event signals
#define XPRE_OFF  331776     // (4,32768) pre-normalization overlap-add
#define BMAX_OFF  462848     // (4,128)  per-block maxes

typedef float v2f __attribute__((ext_vector_type(2)));
typedef float v8f __attribute__((ext_vector_type(8)));

// linear interp of a frame array (align_corners=False semantics, pre-clipped)
__device__ __forceinline__ float frame_lerp(const float* __restrict__ f, int s,
                                            float invScale, int n) {
  float pos = ((float)s + 0.5f) * invScale - 0.5f;
  pos = fminf(fmaxf(pos, 0.0f), (float)(n - 1));
  int lo = (int)pos;                 // pos >= 0, trunc == floor
  int hi = min(lo + 1, n - 1);
  float w = pos - (float)lo;
  return f[lo] * (1.0f - w) + f[hi] * w;
}

// ---------------------------------------------------------------------------
// Kernel 1: dense layers via V_WMMA_F32_16X16X4_F32 (single wave32).
// A = latents padded to 16x16 (rows 8..15 zero); K=16 -> 4 WMMA steps.
// ---------------------------------------------------------------------------
__global__ __launch_bounds__(32) void k_frontend(
    const float* __restrict__ latents, const float* __restrict__ wav,
    const float* __restrict__ Wc, const float* __restrict__ bc,
    const float* __restrict__ We, const float* __restrict__ be,
    const float* __restrict__ Wf, const float* __restrict__ bf,
    float* __restrict__ ws) {
  __shared__ float ldsLogit[BN * 64];

  const int lane = threadIdx.x;
  const int m    = lane & 15;     // M (A) / N (B,C,D) position
  const int half = lane >> 4;     // K sub-pair selector for A/B

  // A tiles: 16x4 f32 layout -> lane holds (K = kt*4 + 2*half + {0,1}) for row m
  v2f A[4];
  for (int kt = 0; kt < 4; ++kt) {
    int k0 = kt * 4 + half * 2;
    float a0 = (m < BN) ? latents[m * MODEL_DIM + k0]     : 0.0f;
    float a1 = (m < BN) ? latents[m * MODEL_DIM + k0 + 1] : 0.0f;
    v2f a = {a0, a1};
    A[kt] = a;
  }

  // ---- choice logits: (16x16)x(16x64) -> 4 N-tiles --------------------------
  for (int nt = 0; nt < 4; ++nt) {
    v8f acc = {0, 0, 0, 0, 0, 0, 0, 0};
    for (int kt = 0; kt < 4; ++kt) {
      int k0 = kt * 4 + half * 2;
      v2f Bv = {Wc[(k0    ) * 64 + nt * 16 + m],
                Wc[(k0 + 1) * 64 + nt * 16 + m]};
      acc = __builtin_amdgcn_wmma_f32_16x16x4_f32(false, A[kt], false, Bv,
                                                  (short)0, acc, false, false);
    }
    if (half == 0) {               // lanes 0..15 hold rows M=0..7 in acc[0..7]
      for (int r = 0; r < BN; ++r)
        ldsLogit[r * 64 + nt * 16 + m] = acc[r] + bc[nt * 16 + m];
    }
  }
  __syncthreads();
  // softmax over the 4 tables (axis k, stride 16)
  for (int i = lane; i < BN * 16; i += 32) {
    int row = i >> 4, j = i & 15;
    float v[NTAB], mx = -INFINITY;
    for (int k = 0; k < NTAB; ++k) {
      v[k] = ldsLogit[row * 64 + k * 16 + j];
      mx = fmaxf(mx, v[k]);
    }
    float sum = 0.0f;
    for (int k = 0; k < NTAB; ++k) { v[k] = __expf(v[k] - mx); sum += v[k]; }
    float inv = 1.0f / sum;
    for (int k = 0; k < NTAB; ++k)
      ws[C16_OFF + row * 64 + k * 16 + j] = v[k] * inv;
  }

  // ---- envelope frames: (16x16)x(16x128) -> 8 N-tiles, abs ------------------
  for (int nt = 0; nt < 8; ++nt) {
    v8f acc = {0, 0, 0, 0, 0, 0, 0, 0};
    for (int kt = 0; kt < 4; ++kt) {
      int k0 = kt * 4 + half * 2;
      v2f Bv = {We[(k0    ) * NFRAMES + nt * 16 + m],
                We[(k0 + 1) * NFRAMES + nt * 16 + m]};
      acc = __builtin_amdgcn_wmma_f32_16x16x4_f32(false, A[kt], false, Bv,
                                                  (short)0, acc, false, false);
    }
    if (half == 0) {
      for (int r = 0; r < BN; ++r)
        ws[ENVF_OFF + r * NFRAMES + nt * 16 + m] =
            fabsf(acc[r] + be[nt * 16 + m]);
    }
  }

  // ---- frequency frames: sigmoid scaled -------------------------------------
  for (int nt = 0; nt < 8; ++nt) {
    v8f acc = {0, 0, 0, 0, 0, 0, 0, 0};
    for (int kt = 0; kt < 4; ++kt) {
      int k0 = kt * 4 + half * 2;
      v2f Bv = {Wf[(k0    ) * NFRAMES + nt * 16 + m],
                Wf[(k0 + 1) * NFRAMES + nt * 16 + m]};
      acc = __builtin_amdgcn_wmma_f32_16x16x4_f32(false, A[kt], false, Bv,
                                                  (short)0, acc, false, false);
    }
    if (half == 0) {
      for (int r = 0; r < BN; ++r) {
        float v = acc[r] + bf[nt * 16 + m];
        ws[FREQF_OFF + r * NFRAMES + nt * 16 + m] =
            0.0009070294784580499f + 0.2f / (1.0f + __expf(-v));
      }
    }
  }

  // ---- per-table max (plain max, matching jnp.max) --------------------------
  if (lane < NTAB) {
    float mx = -INFINITY;
    for (int t = 0; t < TSIZE; ++t) mx = fmaxf(mx, wav[lane * TSIZE + t]);
    ws[WTMAX_OFF + lane] = mx;
  }
}

// ---------------------------------------------------------------------------
// Kernel 1b: mixed16[b][j][t] = sum_k (wav[k][t]/(max_k+1e-8)) * c16[b][k][j]
// (interp along samples commutes with the linear table mix)
// ---------------------------------------------------------------------------
__global__ __launch_bounds__(256) void k_mixed16(const float* __restrict__ wav,
                                                 float* __restrict__ ws) {
  int b = blockIdx.x;
  float wm[NTAB], cj[NTAB];
  for (int k = 0; k < NTAB; ++k) wm[k] = 1.0f / (ws[WTMAX_OFF + k] + 1e-8f);
  for (int i = threadIdx.x; i < 16 * TSIZE; i += 256) {
    int j = i >> 9, t = i & (TSIZE - 1);
    for (int k = 0; k < NTAB; ++k) cj[k] = ws[C16_OFF + b * 64 + k * 16 + j];
    float s = 0.0f;
    for (int k = 0; k < NTAB; ++k) s += wav[k * TSIZE + t] * wm[k] * cj[k];
    ws[MIX_OFF + b * (16 * TSIZE) + i] = s;
  }
}

// ---------------------------------------------------------------------------
// Kernel 1c: per-event 256-sample block sums of freq + exclusive scan.
// ---------------------------------------------------------------------------
__global__ __launch_bounds__(128) void k_prefix(float* __restrict__ ws) {
  __shared__ float bsum[128];
  int b = blockIdx.x, blk = threadIdx.x;
  const float* fF = ws + FREQF_OFF + b * NFRAMES;
  float s = 0.0f;
  for (int q = 0; q < 256; ++q)
    s += frame_lerp(fF, blk * 256 + q, 1.0f / 256.0f, NFRAMES);
  bsum[blk] = s;
  __syncthreads();
  if (blk == 0) {
    float run = 0.0f;
    for (int i = 0; i < 128; ++i) { float t = bsum[i]; bsum[i] = run; run += t; }
  }
  __syncthreads();
  ws[PREF_OFF + b * 128 + blk] = bsum[blk];
}

// ---------------------------------------------------------------------------
// Kernel 2: main streaming kernel. One WG per (event, 256-sample block).
// Each wave emits one sample's 512 Gaussian values as 4 x float4 per lane.
// ---------------------------------------------------------------------------
__global__ __launch_bounds__(256) void k_main(float* __restrict__ ws,
                                              float* __restrict__ out) {
  __shared__ __align__(16) float mlds[16 * TSIZE];   // [j][t], 32KB
  __shared__ float phs[256];
  __shared__ float env[256];
  __shared__ float scan[256];

  const int b   = blockIdx.x >> 7;
  const int blk = blockIdx.x & 127;
  const int tid = threadIdx.x;

  const float* mg = ws + MIX_OFF + (size_t)b * (16 * TSIZE);
  __builtin_prefetch(mg + tid * 4, 0, 0);
  for (int i = tid; i < 16 * TSIZE; i += 256) mlds[i] = mg[i];

  const int s = blk * 256 + tid;
  const float* fF = ws + FREQF_OFF + b * NFRAMES;
  scan[tid] = frame_lerp(fF, s, 1.0f / 256.0f, NFRAMES);
  __syncthreads();
  for (int off = 1; off < 256; off <<= 1) {     // inclusive Hillis-Steele
    float mine  = scan[tid];
    float other = (tid >= off) ? scan[tid - off] : 0.0f;
    __syncthreads();
    scan[tid] = mine + other;
    __syncthreads();
  }
  float ph = scan[tid] + ws[PREF_OFF + b * 128 + blk];
  phs[tid] = ph - floorf(ph);                    // cumsum mod 1
  const float* eF = ws + ENVF_OFF + b * NFRAMES;
  env[tid] = frame_lerp(eF, s, 1.0f / 256.0f, NFRAMES);
  __syncthreads();

  const int wid = tid >> 5, lane = tid & 31;
  float* skb = out + SK_OFF + (size_t)b * S_TOT * TSIZE;
  float* ev  = ws + EV_OFF + (size_t)b * S_TOT;

  for (int si = wid * 32; si < wid * 32 + 32; ++si) {
    const int ss = blk * 256 + si;
    const float p = phs[si];
    float posc = ((float)ss + 0.5f) * (1.0f / 2048.0f) - 0.5f;
    posc = fminf(fmaxf(posc, 0.0f), 15.0f);
    const int lo = (int)posc;
    const int hi = min(lo + 1, 15);
    const float wc = posc - (float)lo;

    float acc = 0.0f;
    float* skrow = skb + (size_t)ss * TSIZE;
    for (int it = 0; it < 4; ++it) {
      const int t0 = it * 128 + lane * 4;
      const float4 mlo = *reinterpret_cast<const float4*>(&mlds[lo * TSIZE + t0]);
      const float4 mhi = *reinterpret_cast<const float4*>(&mlds[hi * TSIZE + t0]);
      float4 g;
      float d;
      d = ((float)(t0 + 0) * (1.0f / 511.0f) - p) * 100.0f;
      g.x = __expf(-0.5f * d * d) * INV_GAUSS;
      d = ((float)(t0 + 1) * (1.0f / 511.0f) - p) * 100.0f;
      g.y = __expf(-0.5f * d * d) * INV_GAUSS;
      d = ((float)(t0 + 2) * (1.0f / 511.0f) - p) * 100.0f;
      g.z = __expf(-0.5f * d * d) * INV_GAUSS;
      d = ((float)(t0 + 3) * (1.0f / 511.0f) - p) * 100.0f;
      g.w = __expf(-0.5f * d * d) * INV_GAUSS;

      acc += g.x * (mlo.x + (mhi.x - mlo.x) * wc);
      acc += g.y * (mlo.y + (mhi.y - mlo.y) * wc);
      acc += g.z * (mlo.z + (mhi.z - mlo.z) * wc);
      acc += g.w * (mlo.w + (mhi.w - mlo.w) * wc);

      *reinterpret_cast<float4*>(&skrow[t0]) = g;   // coalesced b128 store
    }
    for (int off = 16; off; off >>= 1) acc += __shfl_xor(acc, off, 32);
    if (lane == 0) ev[ss] = acc * env[si];
  }
}

// ---------------------------------------------------------------------------
// Kernel 3: overlap-add (gather form, no atomics) + per-block max.
// ---------------------------------------------------------------------------
__global__ __launch_bounds__(256) void k_ola(const int* __restrict__ idx,
                                             float* __restrict__ ws) {
  __shared__ float red[256];
  const int b = blockIdx.x >> 7, blk = blockIdx.x & 127;
  const int p = blk * 256 + threadIdx.x;
  float v = 0.0f;
  for (int e = 0; e < 2; ++e) {
    int off = p - idx[b * 2 + e] * HOP;
    if (off >= 0 && off < S_TOT)
      v += ws[EV_OFF + (size_t)(b * 2 + e) * S_TOT + off];
  }
  ws[XPRE_OFF + (size_t)b * S_TOT + p] = v;
  red[threadIdx.x] = v;
  __syncthreads();
  for (int o = 128; o; o >>= 1) {
    if (threadIdx.x < o)
      red[threadIdx.x] = fmaxf(red[threadIdx.x], red[threadIdx.x + o]);
    __syncthreads();
  }
  if (threadIdx.x == 0) ws[BMAX_OFF + b * 128 + blk] = red[0];
}

// ---------------------------------------------------------------------------
// Kernel 4: reduce maxes, normalize, write final x to d_out.
// ---------------------------------------------------------------------------
__global__ __launch_bounds__(256) void k_norm(float* __restrict__ ws,
                                              float* __restrict__ out) {
  __shared__ float red[128];
  const int b = blockIdx.x;
  if (threadIdx.x < 128) red[threadIdx.x] = ws[BMAX_OFF + b * 128 + threadIdx.x];
  __syncthreads();
  for (int o = 64; o; o >>= 1) {
    if (threadIdx.x < o)
      red[threadIdx.x] = fmaxf(red[threadIdx.x], red[threadIdx.x + o]);
    __syncthreads();
  }
  const float inv = 1.0f / (red[0] + 1e-8f);
  for (int p = threadIdx.x; p < S_TOT; p += 256)
    out[X_OFF + (size_t)b * S_TOT + p] =
        ws[XPRE_OFF + (size_t)b * S_TOT + p] * inv;
}

// ---------------------------------------------------------------------------
extern "C" void kernel_launch(void* const* d_in, const int* in_sizes, int n_in,
                              void* d_out, int out_size, void* d_ws,
                              size_t ws_size, hipStream_t stream) {
  (void)in_sizes; (void)n_in; (void)out_size; (void)ws_size;
  const float* latents = (const float*)d_in[0];
  const float* wav     = (const float*)d_in[1];
  const float* Wc      = (const float*)d_in[2];
  const float* bc      = (const float*)d_in[3];
  const float* We      = (const float*)d_in[4];
  const float* be      = (const float*)d_in[5];
  const float* Wf      = (const float*)d_in[6];
  const float* bf      = (const float*)d_in[7];
  const int*   indices = (const int*)d_in[8];
  float* out = (float*)d_out;
  float* ws  = (float*)d_ws;

  k_frontend<<<1, 32, 0, stream>>>(latents, wav, Wc, bc, We, be, Wf, bf, ws);
  k_mixed16<<<BN, 256, 0, stream>>>(wav, ws);
  k_prefix<<<BN, 128, 0, stream>>>(ws);
  k_main<<<BN * 128, 256, 0, stream>>>(ws, out);
  k_ola<<<BATCH * 128, 256, 0, stream>>>(indices, ws);
  k_norm<<<BATCH, 256, 0, stream>>>(ws, out);
}